// ODEfunc_43679817400336
// MI455X (gfx1250) — compile-verified
//
#include <hip/hip_runtime.h>
#include <hip/hip_bf16.h>
#include <stdint.h>

// Neural-ODE fused step + exact divergence for MI455X (gfx1250).
//   h  = tanh(y @ W1 + t*bt + b1)        [B,512]
//   dy = h @ W2 + b2                     [B,64]
//   div[b] = sum_k (1 - h[b,k]^2) * c_k, c_k = sum_j W1[j,k]*W2[k,j]
// Output = [dy (B*64) | -div (B)] concatenated, fp32.
//
// 4 waves/WG, each wave owns TWO 16-row M-tiles (32 rows) so every LDS
// B-fragment feeds two independent v_wmma chains (halves DS traffic/wmma,
// doubles ILP at the 1-wave/SIMD occupancy forced by the 300 KB LDS tile).

#define D_DIM 64
#define H_DIM 512
#define ROWS_PER_WG 128
#define THREADS 128
#define WAVES 4

typedef float v2f __attribute__((ext_vector_type(2)));
typedef float v8f __attribute__((ext_vector_type(8)));
typedef int   v4i __attribute__((ext_vector_type(4)));

#if __has_builtin(__builtin_amdgcn_global_load_async_to_lds_b128) && \
    __has_builtin(__builtin_amdgcn_s_wait_asynccnt)
#define HAVE_ASYNC_LDS 1
typedef __attribute__((address_space(1))) v4i* g4p_t;   // global int4*
typedef __attribute__((address_space(3))) v4i* l4p_t;   // LDS int4*
#endif

#if __has_builtin(__builtin_amdgcn_tanhf)
#define TANHF(x) __builtin_amdgcn_tanhf(x)
#else
#define TANHF(x) tanhf(x)
#endif

__device__ __forceinline__ void stage_to_lds(float* dst, const float* __restrict__ src,
                                             int nfloats, int tid) {
  const int nvec = nfloats >> 2;  // 16B units
#if defined(HAVE_ASYNC_LDS)
  for (int i = tid; i < nvec; i += THREADS) {
    __builtin_amdgcn_global_load_async_to_lds_b128(
        (g4p_t)(src + 4 * i), (l4p_t)(dst + 4 * i), 0, 0);
  }
#else
  for (int i = tid; i < nvec; i += THREADS) {
    ((float4*)dst)[i] = ((const float4*)src)[i];
  }
#endif
}

__global__ __launch_bounds__(THREADS)
void ode_fused_wmma_kernel(const float* __restrict__ Y, const float* __restrict__ Tsc,
                           const float* __restrict__ W1g, const float* __restrict__ Btg,
                           const float* __restrict__ B1g, const float* __restrict__ W2g,
                           const float* __restrict__ B2g, float* __restrict__ Out,
                           int Brows) {
  extern __shared__ float smem[];
  float* sW1   = smem;            // [64][512]  32768 f  (128 KB)
  float* sW2   = smem + 32768;    // [512][64]  32768 f  (128 KB)
  float* sY    = smem + 65536;    // [128][64]   8192 f  ( 32 KB)
  float* sBias = smem + 73728;    // [512]  t*bt + b1
  float* sC    = smem + 74240;    // [512]  c_k
  float* sB2   = smem + 74752;    // [64]
  float* sHt   = smem + 74816;    // 4 waves * 2 tiles * [16][16] h scratch

  const int tid  = threadIdx.x;
  const int wave = tid >> 5;
  const int lane = tid & 31;
  const int half = lane >> 4;     // 0: lanes 0-15, 1: lanes 16-31
  const int lq   = lane & 15;
  const int rowBase = blockIdx.x * ROWS_PER_WG;

  // ---- Stage weights + batch tile into LDS (async DMA path on CDNA5) ----
  stage_to_lds(sW1, W1g, D_DIM * H_DIM, tid);
  stage_to_lds(sW2, W2g, H_DIM * D_DIM, tid);
  stage_to_lds(sY, Y + (size_t)rowBase * D_DIM, ROWS_PER_WG * D_DIM, tid);

  const float t = Tsc[0];
  for (int h = tid; h < H_DIM; h += THREADS) sBias[h] = t * Btg[h] + B1g[h];
  if (tid < D_DIM) sB2[tid] = B2g[tid];

#if defined(HAVE_ASYNC_LDS)
  __builtin_amdgcn_s_wait_asynccnt(0);
#endif
  __syncthreads();

  // ---- c_k = sum_j W1[j,k] * W2[k,j]  (tiny: 512*64 MACs per WG) ----
  for (int h = tid; h < H_DIM; h += THREADS) {
    float c = 0.f;
#pragma unroll 16
    for (int j = 0; j < D_DIM; ++j) c += sW1[j * H_DIM + h] * sW2[h * D_DIM + j];
    sC[h] = c;
  }
  __syncthreads();

  // ---- Per-wave: two 16-row M-tiles (rows waveRow .. waveRow+31) ----
  const int waveRow = wave * 32;

  // A fragments for GEMM1 (y tiles, K=64 -> 16 chunks of 16x4 each).
  // A layout: lanes 0-15 hold M=lane, K={0,1}; lanes 16-31 hold M=lane-16, K={2,3}.
  v2f a1[2][16];
#pragma unroll
  for (int m = 0; m < 2; ++m)
#pragma unroll
    for (int kk = 0; kk < 16; ++kk)
      a1[m][kk] = *(const v2f*)&sY[(waveRow + m * 16 + lq) * D_DIM + kk * 4 + half * 2];

  // Persistent GEMM2 accumulators (D=64 -> 4 N-tiles per M-tile), seeded with b2.
  v8f acc2[2][4];
#pragma unroll
  for (int nt = 0; nt < 4; ++nt) {
    float bv = sB2[nt * 16 + lq];
    acc2[0][nt] = (v8f){bv, bv, bv, bv, bv, bv, bv, bv};
    acc2[1][nt] = acc2[0][nt];
  }
  float divAcc[2][8];
#pragma unroll
  for (int m = 0; m < 2; ++m)
#pragma unroll
    for (int r = 0; r < 8; ++r) divAcc[m][r] = 0.f;

  float* myHt = sHt + wave * 512;  // two 16x16 tiles

  for (int n = 0; n < 32; ++n) {  // hidden tiles of 16
    // z tiles = y @ W1 + bias ; C/D layout: vgpr r -> M = r + 8*half, N = 16n+lq
    float zb = sBias[n * 16 + lq];
    v8f z0 = (v8f){zb, zb, zb, zb, zb, zb, zb, zb};
    v8f z1 = z0;
#pragma unroll
    for (int kk = 0; kk < 16; ++kk) {
      // B 4x16 layout: vgpr v -> K = v + 2*half, N = lq   (shared by both M-tiles)
      v2f b;
      const float* bp = &sW1[(kk * 4 + half * 2) * H_DIM + n * 16 + lq];
      b.x = bp[0];
      b.y = bp[H_DIM];
      z0 = __builtin_amdgcn_wmma_f32_16x16x4_f32(false, a1[0][kk], false, b,
                                                 (short)0, z0, false, false);
      z1 = __builtin_amdgcn_wmma_f32_16x16x4_f32(false, a1[1][kk], false, b,
                                                 (short)0, z1, false, false);
    }

    // tanh (gfx1250 v_tanh_f32), divergence weights, stash h tiles to LDS.
    const float cv = sC[n * 16 + lq];
#pragma unroll
    for (int r = 0; r < 8; ++r) {
      float h0 = TANHF(z0[r]);
      float h1 = TANHF(z1[r]);
      divAcc[0][r] += (1.f - h0 * h0) * cv;
      divAcc[1][r] += (1.f - h1 * h1) * cv;
      myHt[(r + half * 8) * 16 + lq]       = h0;
      myHt[256 + (r + half * 8) * 16 + lq] = h1;
    }
    // Same-wave cross-lane RAW through LDS: DS ops are in-order per wave,
    // just drain the DS counter before re-reading as A fragments.
    asm volatile("s_wait_dscnt 0" ::: "memory");

    // GEMM2 partial: this h tile contributes K = 16n .. 16n+15 (4 chunks of 4).
#pragma unroll
    for (int kk2 = 0; kk2 < 4; ++kk2) {
      v2f a2_0 = *(const v2f*)&myHt[lq * 16 + kk2 * 4 + half * 2];
      v2f a2_1 = *(const v2f*)&myHt[256 + lq * 16 + kk2 * 4 + half * 2];
#pragma unroll
      for (int nt = 0; nt < 4; ++nt) {
        v2f b;  // shared by both M-tiles
        const float* bp = &sW2[(n * 16 + kk2 * 4 + half * 2) * D_DIM + nt * 16 + lq];
        b.x = bp[0];
        b.y = bp[D_DIM];
        acc2[0][nt] = __builtin_amdgcn_wmma_f32_16x16x4_f32(false, a2_0, false, b,
                                                            (short)0, acc2[0][nt], false, false);
        acc2[1][nt] = __builtin_amdgcn_wmma_f32_16x16x4_f32(false, a2_1, false, b,
                                                            (short)0, acc2[1][nt], false, false);
      }
    }
  }

  // ---- Write dy ----
#pragma unroll
  for (int m = 0; m < 2; ++m) {
    const int gRow = rowBase + waveRow + m * 16;
#pragma unroll
    for (int nt = 0; nt < 4; ++nt) {
#pragma unroll
      for (int r = 0; r < 8; ++r) {
        Out[(size_t)(gRow + r + half * 8) * D_DIM + nt * 16 + lq] = acc2[m][nt][r];
      }
    }
  }

  // ---- Reduce divergence across the 16-lane N-group, write -div ----
  float* divOut = Out + (size_t)Brows * D_DIM;
#pragma unroll
  for (int m = 0; m < 2; ++m) {
#pragma unroll
    for (int r = 0; r < 8; ++r) {
      float v = divAcc[m][r];
#pragma unroll
      for (int s = 1; s < 16; s <<= 1) v += __shfl_xor(v, s, 32);
      divAcc[m][r] = v;
    }
    if (lq == 0) {  // lanes 0 (rows 0-7) and 16 (rows 8-15)
      const int gRow = rowBase + waveRow + m * 16;
#pragma unroll
      for (int r = 0; r < 8; ++r) divOut[gRow + r + half * 8] = -divAcc[m][r];
    }
  }
}

extern "C" void kernel_launch(void* const* d_in, const int* in_sizes, int n_in,
                              void* d_out, int out_size, void* d_ws, size_t ws_size,
                              hipStream_t stream) {
  (void)n_in; (void)out_size; (void)d_ws; (void)ws_size;
  const float* Y  = (const float*)d_in[0];
  const float* T  = (const float*)d_in[1];
  const float* W1 = (const float*)d_in[2];
  const float* Bt = (const float*)d_in[3];
  const float* B1 = (const float*)d_in[4];
  const float* W2 = (const float*)d_in[5];
  const float* B2 = (const float*)d_in[6];
  float* Out = (float*)d_out;

  const int Brows = in_sizes[0] / D_DIM;          // 32768
  const int grid  = Brows / ROWS_PER_WG;          // 256 workgroups
  const size_t shmem = 76864u * sizeof(float);    // ~300 KB of the 320 KB WGP LDS

  ode_fused_wmma_kernel<<<grid, THREADS, shmem, stream>>>(Y, T, W1, Bt, B1, W2, B2,
                                                          Out, Brows);
}